// LiquidNet_10307921510844
// MI455X (gfx1250) — compile-verified
//
#include <hip/hip_runtime.h>
#include <hip/hip_bf16.h>
#include <stdint.h>

#define BATCH   64
#define FIN     256
#define HU      768
#define BH      (BATCH * HU)
#define UNFOLDS 6
#define LOG2E   1.4426950408889634f

#define TB    8                 // batch rows per block (== blockDim.y, one wave each)
#define SPLIT 4                 // i-reduction split for the step kernel
#define ILEN  (HU / SPLIT)      // 192

// ---- CDNA5 async global->LDS copy (ASYNCcnt path, gfx1250) -----------------
// GVS form: global_load_async_to_lds_b128 vDstLds, vOff32, sBase64
__device__ __forceinline__ void async_copy_b128(uint32_t lds_byte,
                                                const float* base,
                                                uint32_t voff_bytes) {
    asm volatile("global_load_async_to_lds_b128 %0, %1, %2"
                 :
                 : "v"(lds_byte), "v"(voff_bytes), "s"(base)
                 : "memory");
}
__device__ __forceinline__ void async_wait0() {
    asm volatile("s_wait_asynccnt 0" ::: "memory");
}
__device__ __forceinline__ uint32_t lds_off_u32(const void* p) {
    return (uint32_t)(uintptr_t)p;   // flat LDS addr: low 32 bits = LDS offset
}
__device__ __forceinline__ float sig_from_arg(float a) {
    // a = sigma*log2e*(mu - v)  ->  sigmoid(sigma*(v-mu)) = 1/(1+2^a)
    return __builtin_amdgcn_rcpf(1.0f + __builtin_amdgcn_exp2f(a));
}

// ---------------------------------------------------------------------------
// HOT KERNEL (placed first so the disasm snippet shows its inner loop)
// Tier-2 step partial: split-K over SPLIT i-ranges, 4 h per thread
// per element: 1 v_fma (arg) + v_exp_f32 + v_add + v_rcp_f32 + 2 v_fma (acc)
// ---------------------------------------------------------------------------
__global__ __launch_bounds__(256) void ltc_step_partial(
    const float* __restrict__ vpre,     // B x H
    const float* __restrict__ sgL, const float* __restrict__ c,
    const float* __restrict__ W,   const float* __restrict__ We,   // H x H
    float* __restrict__ pnum, float* __restrict__ pden)            // SPLIT x B x H
{
    __shared__ __align__(16) float sv[TB * ILEN];
    const int tx = threadIdx.x, ty = threadIdx.y;
    const int hb = blockIdx.x * 128;
    const int bb = blockIdx.y * TB;
    const int i0 = blockIdx.z * ILEN;

    // Static async staging: wave ty owns batch row (bb+ty); 48 b128 chunks/row.
    const uint32_t sv_base = lds_off_u32(&sv[0]) + (uint32_t)(ty * ILEN * 4);
    const uint32_t gbase   = (uint32_t)(((bb + ty) * HU + i0) * 4);
    async_copy_b128(sv_base + (uint32_t)(tx * 16), vpre, gbase + (uint32_t)(tx * 16));
    if (tx < 16)
        async_copy_b128(sv_base + (uint32_t)((32 + tx) * 16), vpre,
                        gbase + (uint32_t)((32 + tx) * 16));
    async_wait0();
    __syncthreads();

    const int h0 = hb + tx * 4;
    const int b  = bb + ty;
    float4 num = {0,0,0,0}, den = {0,0,0,0};
    const float* vr = &sv[ty * ILEN];

    for (int i = 0; i < ILEN; i += 4) {
        float4 vv = *(const float4*)&vr[i];             // ds_load_b128
#pragma unroll
        for (int u = 0; u < 4; ++u) {
            const int idx = (i0 + i + u) * HU + h0;
            float4 g4  = *(const float4*)&sgL[idx];
            float4 c4  = *(const float4*)&c[idx];
            float4 w4  = *(const float4*)&W[idx];
            float4 we4 = *(const float4*)&We[idx];
            float  v   = (&vv.x)[u];
            float s0 = sig_from_arg(fmaf(-g4.x, v, c4.x));
            float s1 = sig_from_arg(fmaf(-g4.y, v, c4.y));
            float s2 = sig_from_arg(fmaf(-g4.z, v, c4.z));
            float s3 = sig_from_arg(fmaf(-g4.w, v, c4.w));
            den.x = fmaf(w4.x, s0, den.x);  den.y = fmaf(w4.y, s1, den.y);
            den.z = fmaf(w4.z, s2, den.z);  den.w = fmaf(w4.w, s3, den.w);
            num.x = fmaf(we4.x, s0, num.x); num.y = fmaf(we4.y, s1, num.y);
            num.z = fmaf(we4.z, s2, num.z); num.w = fmaf(we4.w, s3, num.w);
        }
    }
    const int o = blockIdx.z * BH + b * HU + h0;
    *(float4*)&pnum[o] = num;
    *(float4*)&pden[o] = den;
}

// ---------------------------------------------------------------------------
// Tier-2 precompute: sgL = sigma*log2e ; c = sgL*mu ; We = W*erev
// ---------------------------------------------------------------------------
__global__ __launch_bounds__(256) void ltc_precompute(
    const float4* __restrict__ sigma, const float4* __restrict__ mu,
    const float4* __restrict__ W,     const float4* __restrict__ erev,
    float4* __restrict__ sgL, float4* __restrict__ c, float4* __restrict__ We,
    int n4)
{
    int i = blockIdx.x * 256 + threadIdx.x;
    if (i >= n4) return;
    float4 s4 = sigma[i], m4 = mu[i], w4 = W[i], e4 = erev[i];
    float4 g, cc, we;
    g.x = s4.x * LOG2E;  g.y = s4.y * LOG2E;  g.z = s4.z * LOG2E;  g.w = s4.w * LOG2E;
    cc.x = g.x * m4.x;   cc.y = g.y * m4.y;   cc.z = g.z * m4.z;   cc.w = g.w * m4.w;
    we.x = w4.x * e4.x;  we.y = w4.y * e4.y;  we.z = w4.z * e4.z;  we.w = w4.w * e4.w;
    sgL[i] = g;  c[i] = cc;  We[i] = we;
}

// ---------------------------------------------------------------------------
// Tier-2 sensory pass (precomputed arrays), 4 h per thread
// ---------------------------------------------------------------------------
__global__ __launch_bounds__(256) void ltc_sensory_pre(
    const float* __restrict__ inputs,   // B x F
    const float* __restrict__ input_w, const float* __restrict__ input_b,
    const float* __restrict__ sgL,      // F x H
    const float* __restrict__ c,        // F x H
    const float* __restrict__ W,        // F x H (original sensory_W)
    const float* __restrict__ We,       // F x H
    float* __restrict__ wnum_s, float* __restrict__ wden_s)
{
    __shared__ __align__(16) float sx[TB * FIN];
    const int tx = threadIdx.x, ty = threadIdx.y, tid = ty * 32 + tx;
    const int hb = blockIdx.x * 128;       // 32 threads * 4 h
    const int bb = blockIdx.y * TB;

    // Static async staging: wave ty owns batch row (bb+ty); 64 b128 chunks/row.
    const uint32_t sx_base = lds_off_u32(&sx[0]) + (uint32_t)(ty * FIN * 4);
    const uint32_t gbase   = (uint32_t)(((bb + ty) * FIN) * 4);
    async_copy_b128(sx_base + (uint32_t)(tx * 16), inputs, gbase + (uint32_t)(tx * 16));
    async_copy_b128(sx_base + (uint32_t)((32 + tx) * 16), inputs,
                    gbase + (uint32_t)((32 + tx) * 16));
    async_wait0();
    __syncthreads();
    for (int k = tid; k < TB * FIN; k += 256) {       // affine x = x*w+b
        int f = k & (FIN - 1);
        sx[k] = sx[k] * input_w[f] + input_b[f];
    }
    __syncthreads();

    const int h0 = hb + tx * 4;
    const int b  = bb + ty;
    float4 num = {0,0,0,0}, den = {0,0,0,0};
    const float* xr = &sx[ty * FIN];

    for (int f = 0; f < FIN; f += 4) {
        float4 xv = *(const float4*)&xr[f];
#pragma unroll
        for (int u = 0; u < 4; ++u) {
            const int idx = (f + u) * HU + h0;
            float4 g4  = *(const float4*)&sgL[idx];
            float4 c4  = *(const float4*)&c[idx];
            float4 w4  = *(const float4*)&W[idx];
            float4 we4 = *(const float4*)&We[idx];
            float  x   = (&xv.x)[u];
            float s0 = sig_from_arg(fmaf(-g4.x, x, c4.x));
            float s1 = sig_from_arg(fmaf(-g4.y, x, c4.y));
            float s2 = sig_from_arg(fmaf(-g4.z, x, c4.z));
            float s3 = sig_from_arg(fmaf(-g4.w, x, c4.w));
            den.x = fmaf(w4.x, s0, den.x);  den.y = fmaf(w4.y, s1, den.y);
            den.z = fmaf(w4.z, s2, den.z);  den.w = fmaf(w4.w, s3, den.w);
            num.x = fmaf(we4.x, s0, num.x); num.y = fmaf(we4.y, s1, num.y);
            num.z = fmaf(we4.z, s2, num.z); num.w = fmaf(we4.w, s3, num.w);
        }
    }
    *(float4*)&wnum_s[b * HU + h0] = num;
    *(float4*)&wden_s[b * HU + h0] = den;
}

// ---------------------------------------------------------------------------
// Tier-2 combine: fixed-order partial sum + leak + divide (deterministic)
// ---------------------------------------------------------------------------
__global__ __launch_bounds__(256) void ltc_combine(
    const float* __restrict__ vpre,
    const float* __restrict__ pnum, const float* __restrict__ pden,
    const float* __restrict__ wnum_s, const float* __restrict__ wden_s,
    const float* __restrict__ cm_t, const float* __restrict__ gleak,
    const float* __restrict__ vleak,
    float* __restrict__ vnext, float* __restrict__ dout)
{
    const int e = (blockIdx.x * 256 + threadIdx.x) * 4;   // BH/4 threads
    const int h = e % HU;
    float4 v4  = *(const float4*)&vpre[e];
    float4 ns  = *(const float4*)&wnum_s[e];
    float4 ds  = *(const float4*)&wden_s[e];
    float4 cm4 = *(const float4*)&cm_t[h];
    float4 gl4 = *(const float4*)&gleak[h];
    float4 vl4 = *(const float4*)&vleak[h];

    float4 n0 = *(const float4*)&pnum[e];
    float4 n1 = *(const float4*)&pnum[BH + e];
    float4 n2 = *(const float4*)&pnum[2 * BH + e];
    float4 n3 = *(const float4*)&pnum[3 * BH + e];
    float4 d0 = *(const float4*)&pden[e];
    float4 d1 = *(const float4*)&pden[BH + e];
    float4 d2 = *(const float4*)&pden[2 * BH + e];
    float4 d3 = *(const float4*)&pden[3 * BH + e];

    float4 r;
    {
        float num, den;
        num = n0.x+n1.x+n2.x+n3.x + ns.x + cm4.x*v4.x + gl4.x*vl4.x;
        den = d0.x+d1.x+d2.x+d3.x + ds.x + cm4.x + gl4.x;               r.x = num/den;
        num = n0.y+n1.y+n2.y+n3.y + ns.y + cm4.y*v4.y + gl4.y*vl4.y;
        den = d0.y+d1.y+d2.y+d3.y + ds.y + cm4.y + gl4.y;               r.y = num/den;
        num = n0.z+n1.z+n2.z+n3.z + ns.z + cm4.z*v4.z + gl4.z*vl4.z;
        den = d0.z+d1.z+d2.z+d3.z + ds.z + cm4.z + gl4.z;               r.z = num/den;
        num = n0.w+n1.w+n2.w+n3.w + ns.w + cm4.w*v4.w + gl4.w*vl4.w;
        den = d0.w+d1.w+d2.w+d3.w + ds.w + cm4.w + gl4.w;               r.w = num/den;
    }
    *(float4*)&vnext[e] = r;
    if (dout) {
        *(float4*)&dout[e] = r;
        *(float4*)&dout[BH + e] = r;
    }
}

// ---------------------------------------------------------------------------
// Tier-0 fallback (small ws): on-the-fly params, full-i reduction, 2 h/thread
// ---------------------------------------------------------------------------
__global__ __launch_bounds__(256) void ltc_sensory_fly(
    const float* __restrict__ inputs, const float* __restrict__ input_w,
    const float* __restrict__ input_b,
    const float* __restrict__ smu, const float* __restrict__ ssigma,
    const float* __restrict__ sW,  const float* __restrict__ serev,
    float* __restrict__ wnum_s, float* __restrict__ wden_s)
{
    __shared__ __align__(16) float sx[TB * FIN];
    const int tx = threadIdx.x, ty = threadIdx.y, tid = ty * 32 + tx;
    const int hb = blockIdx.x * 64, bb = blockIdx.y * TB;

    const uint32_t sx_base = lds_off_u32(&sx[0]) + (uint32_t)(ty * FIN * 4);
    const uint32_t gbase   = (uint32_t)(((bb + ty) * FIN) * 4);
    async_copy_b128(sx_base + (uint32_t)(tx * 16), inputs, gbase + (uint32_t)(tx * 16));
    async_copy_b128(sx_base + (uint32_t)((32 + tx) * 16), inputs,
                    gbase + (uint32_t)((32 + tx) * 16));
    async_wait0();
    __syncthreads();
    for (int k = tid; k < TB * FIN; k += 256) {
        int f = k & (FIN - 1);
        sx[k] = sx[k] * input_w[f] + input_b[f];
    }
    __syncthreads();

    const int h0 = hb + tx * 2, b = bb + ty;
    float num0 = 0.f, num1 = 0.f, den0 = 0.f, den1 = 0.f;
    const float* xr = &sx[ty * FIN];
    for (int i = 0; i < FIN; i += 4) {
        float4 xv = *(const float4*)&xr[i];
#pragma unroll
        for (int u = 0; u < 4; ++u) {
            const int idx = (i + u) * HU + h0;
            float2 mu2 = *(const float2*)&smu[idx];
            float2 sg2 = *(const float2*)&ssigma[idx];
            float2 w2  = *(const float2*)&sW[idx];
            float2 e2  = *(const float2*)&serev[idx];
            float  x   = (&xv.x)[u];
            float s0 = sig_from_arg(sg2.x * LOG2E * (mu2.x - x));
            float s1 = sig_from_arg(sg2.y * LOG2E * (mu2.y - x));
            den0 += w2.x * s0;          den1 += w2.y * s1;
            num0 += (w2.x * e2.x) * s0; num1 += (w2.y * e2.y) * s1;
        }
    }
    const int o = b * HU + h0;
    wnum_s[o] = num0;  wnum_s[o + 1] = num1;
    wden_s[o] = den0;  wden_s[o + 1] = den1;
}

__global__ __launch_bounds__(256) void ltc_step_fly(
    const float* __restrict__ vpre,
    const float* __restrict__ mu, const float* __restrict__ sigma,
    const float* __restrict__ W,  const float* __restrict__ erev,
    const float* __restrict__ wnum_s, const float* __restrict__ wden_s,
    const float* __restrict__ cm_t, const float* __restrict__ gleak,
    const float* __restrict__ vleak,
    float* __restrict__ vnext, float* __restrict__ dout)
{
    __shared__ __align__(16) float sv[TB * HU];
    const int tx = threadIdx.x, ty = threadIdx.y;
    const int hb = blockIdx.x * 64, bb = blockIdx.y * TB;

    // Static async staging: wave ty owns batch row (bb+ty); 192 b128 chunks/row.
    const uint32_t sv_base = lds_off_u32(&sv[0]) + (uint32_t)(ty * HU * 4);
    const uint32_t gbase   = (uint32_t)(((bb + ty) * HU) * 4);
#pragma unroll
    for (int j = 0; j < 6; ++j)
        async_copy_b128(sv_base + (uint32_t)((j * 32 + tx) * 16), vpre,
                        gbase + (uint32_t)((j * 32 + tx) * 16));
    async_wait0();
    __syncthreads();

    const int h0 = hb + tx * 2, b = bb + ty;
    float num0 = 0.f, num1 = 0.f, den0 = 0.f, den1 = 0.f;
    const float* vr = &sv[ty * HU];
    for (int i = 0; i < HU; i += 4) {
        float4 vv = *(const float4*)&vr[i];
#pragma unroll
        for (int u = 0; u < 4; ++u) {
            const int idx = (i + u) * HU + h0;
            float2 mu2 = *(const float2*)&mu[idx];
            float2 sg2 = *(const float2*)&sigma[idx];
            float2 w2  = *(const float2*)&W[idx];
            float2 e2  = *(const float2*)&erev[idx];
            float  v   = (&vv.x)[u];
            float s0 = sig_from_arg(sg2.x * LOG2E * (mu2.x - v));
            float s1 = sig_from_arg(sg2.y * LOG2E * (mu2.y - v));
            den0 += w2.x * s0;          den1 += w2.y * s1;
            num0 += (w2.x * e2.x) * s0; num1 += (w2.y * e2.y) * s1;
        }
    }
    const int o = b * HU + h0;
    float vp0 = vr[h0], vp1 = vr[h0 + 1];
    float cm0 = cm_t[h0],  cm1 = cm_t[h0 + 1];
    float gl0 = gleak[h0], gl1 = gleak[h0 + 1];
    float vl0 = vleak[h0], vl1 = vleak[h0 + 1];
    float n0 = cm0 * vp0 + gl0 * vl0 + num0 + wnum_s[o];
    float n1 = cm1 * vp1 + gl1 * vl1 + num1 + wnum_s[o + 1];
    float d0 = cm0 + gl0 + den0 + wden_s[o];
    float d1 = cm1 + gl1 + den1 + wden_s[o + 1];
    float r0 = n0 / d0, r1 = n1 / d1;
    vnext[o] = r0;  vnext[o + 1] = r1;
    if (dout) {
        dout[o] = r0;              dout[o + 1] = r1;
        dout[BH + o] = r0;         dout[BH + o + 1] = r1;
    }
}

// ---------------------------------------------------------------------------
extern "C" void kernel_launch(void* const* d_in, const int* in_sizes, int n_in,
                              void* d_out, int out_size, void* d_ws, size_t ws_size,
                              hipStream_t stream) {
    (void)in_sizes; (void)n_in; (void)out_size;

    const float* inputs  = (const float*)d_in[0];
    const float* state   = (const float*)d_in[1];
    const float* input_w = (const float*)d_in[2];
    const float* input_b = (const float*)d_in[3];
    const float* smu     = (const float*)d_in[4];
    const float* ssigma  = (const float*)d_in[5];
    const float* sW      = (const float*)d_in[6];
    const float* serev   = (const float*)d_in[7];
    const float* mu      = (const float*)d_in[8];
    const float* sigma   = (const float*)d_in[9];
    const float* W       = (const float*)d_in[10];
    const float* erev    = (const float*)d_in[11];
    const float* vleak   = (const float*)d_in[12];
    const float* gleak   = (const float*)d_in[13];
    const float* cm_t    = (const float*)d_in[14];

    float* ws = (float*)d_ws;
    const size_t FH = (size_t)FIN * HU;        // 196608
    const size_t HH = (size_t)HU * HU;         // 589824
    const size_t tier2_floats = 3*FH + 3*HH + 2*BH + 2*SPLIT*BH + 2*BH;

    dim3 blk(32, 8);

    if (ws_size >= tier2_floats * sizeof(float)) {
        // ---- Tier 2: precompute + split-K ----
        float* s_sgL  = ws;
        float* s_c    = s_sgL + FH;
        float* s_We   = s_c   + FH;
        float* r_sgL  = s_We  + FH;
        float* r_c    = r_sgL + HH;
        float* r_We   = r_c   + HH;
        float* wnum_s = r_We  + HH;
        float* wden_s = wnum_s + BH;
        float* pnum   = wden_s + BH;
        float* pden   = pnum + SPLIT * BH;
        float* vA     = pden + SPLIT * BH;
        float* vB     = vA + BH;

        ltc_precompute<<<(int)(FH/4/256), 256, 0, stream>>>(
            (const float4*)ssigma, (const float4*)smu,
            (const float4*)sW, (const float4*)serev,
            (float4*)s_sgL, (float4*)s_c, (float4*)s_We, (int)(FH/4));
        ltc_precompute<<<(int)(HH/4/256), 256, 0, stream>>>(
            (const float4*)sigma, (const float4*)mu,
            (const float4*)W, (const float4*)erev,
            (float4*)r_sgL, (float4*)r_c, (float4*)r_We, (int)(HH/4));

        ltc_sensory_pre<<<dim3(HU/128, BATCH/TB), blk, 0, stream>>>(
            inputs, input_w, input_b, s_sgL, s_c, sW, s_We, wnum_s, wden_s);

        const float* src = state;
        for (int u = 0; u < UNFOLDS; ++u) {
            float* dst = (u & 1) ? vB : vA;
            float* dd  = (u == UNFOLDS - 1) ? (float*)d_out : nullptr;
            ltc_step_partial<<<dim3(HU/128, BATCH/TB, SPLIT), blk, 0, stream>>>(
                src, r_sgL, r_c, W, r_We, pnum, pden);
            ltc_combine<<<BH/4/256, 256, 0, stream>>>(
                src, pnum, pden, wnum_s, wden_s, cm_t, gleak, vleak, dst, dd);
            src = dst;
        }
    } else {
        // ---- Tier 0 fallback: on-the-fly params, monolithic step ----
        float* wnum_s = ws;
        float* wden_s = wnum_s + BH;
        float* vA     = wden_s + BH;
        float* vB     = vA + BH;

        ltc_sensory_fly<<<dim3(HU/64, BATCH/TB), blk, 0, stream>>>(
            inputs, input_w, input_b, smu, ssigma, sW, serev, wnum_s, wden_s);

        const float* src = state;
        for (int u = 0; u < UNFOLDS; ++u) {
            float* dst = (u & 1) ? vB : vA;
            float* dd  = (u == UNFOLDS - 1) ? (float*)d_out : nullptr;
            ltc_step_fly<<<dim3(HU/64, BATCH/TB), blk, 0, stream>>>(
                src, mu, sigma, W, erev, wnum_s, wden_s,
                cm_t, gleak, vleak, dst, dd);
            src = dst;
        }
    }
}